// GCNMax_edge_70609262346359
// MI455X (gfx1250) — compile-verified
//
#include <hip/hip_runtime.h>

typedef __attribute__((ext_vector_type(2))) float v2f;
typedef __attribute__((ext_vector_type(8))) float v8f;

#define FEAT   128   // IN_FEATURES == PRINT_LENGTH == 128
#define SPLIT  12    // 128 // 10
#define TILE_M 16

// ---------------------------------------------------------------------------
// v_s = r_s @ W   via V_WMMA_F32_16X16X4_F32 (exact fp32 path; pipeline is
// HBM-bound so the low-rate f32 WMMA is free). 256 threads = 8 wave32s.
// Each block computes a 16-row stripe; wave w owns N-tile [16w, 16w+16).
// W (64 KB) is staged in LDS once per block and feeds B fragments.
// ---------------------------------------------------------------------------
__global__ void __launch_bounds__(256) gcn_gemm_wmma(
    const float* __restrict__ A,    // [n_nodes, 128]
    const float* __restrict__ W,    // [128, 128]
    float* __restrict__ Vs,         // [n_nodes, 128]
    int n_nodes)
{
    __shared__ float lW[FEAT * FEAT];   // 64 KB of the 320 KB WGP LDS

    // Cooperative 128-bit staging of W into LDS.
    {
        const float4* src = (const float4*)W;
        float4*       dst = (float4*)lW;
        for (int i = threadIdx.x; i < FEAT * FEAT / 4; i += 256) dst[i] = src[i];
    }
    __syncthreads();

    const int lane    = threadIdx.x & 31;
    const int wave    = threadIdx.x >> 5;        // 0..7  -> N tile
    const int rowBase = blockIdx.x * TILE_M;
    const int m       = lane & 15;
    const int koff    = (lane >> 4) * 2;         // f32 A layout: lanes 0-15 K=0,1; 16-31 K=2,3
    const int nbase   = wave * 16 + m;           // B/D column for this lane

    int arow = rowBase + m;
    if (arow >= n_nodes) arow = n_nodes - 1;     // clamp (stores are guarded)
    const float* Arow = A + (size_t)arow * FEAT;

    v8f acc = {};
#pragma unroll 8
    for (int k = 0; k < FEAT; k += 4) {
        // A fragment: 2 consecutive fp32 (8B-aligned) -> global_load_b64
        v2f a = *(const v2f*)(Arow + k + koff);
        // B fragment from LDS (row-major W): mirrored K layout across half-waves
        v2f b;
        b.x = lW[(k + koff)     * FEAT + nbase];
        b.y = lW[(k + koff + 1) * FEAT + nbase];
        acc = __builtin_amdgcn_wmma_f32_16x16x4_f32(
            /*neg_a=*/false, a, /*neg_b=*/false, b,
            /*c_mod=*/(short)0, acc, /*reuse_a=*/false, /*reuse_b=*/false);
    }

    // D layout: VGPR v -> M = v (lanes 0-15) or v+8 (lanes 16-31), N = lane%16
    // Tile-level guard: 50000 % 16 == 0 so every block takes the straight-line
    // path (clause of 8 global_store_b32, no per-element exec juggling).
    const int moff = (lane >> 4) * 8;
    float* drow = Vs + (size_t)(rowBase + moff) * FEAT + nbase;
    if (rowBase + TILE_M <= n_nodes) {
#pragma unroll
        for (int v = 0; v < 8; ++v) drow[(size_t)v * FEAT] = acc[v];
    } else {
#pragma unroll
        for (int v = 0; v < 8; ++v)
            if (rowBase + moff + v < n_nodes) drow[(size_t)v * FEAT] = acc[v];
    }
}

// ---------------------------------------------------------------------------
// agg[row[e], c] += v_s[col[e], c]  for c in [0, SPLIT) only — the reference
// discards agg columns >= split, so we never move the other 116 columns.
// One thread per (edge, c) pair; relaxed agent-scope float atomics.
// ---------------------------------------------------------------------------
__global__ void __launch_bounds__(256) gcn_scatter(
    const float* __restrict__ Vs,    // [n_nodes, 128]
    const int*   __restrict__ ei,    // [2, n_edges] flattened
    float*       __restrict__ agg,   // [n_nodes, SPLIT]
    int n_edges)
{
    int i = blockIdx.x * blockDim.x + threadIdx.x;
    int e = i / SPLIT;
    int c = i - e * SPLIT;
    if (e >= n_edges) return;
    int r  = ei[e];             // destination node
    int cl = ei[n_edges + e];   // source node
    float v = Vs[(size_t)cl * FEAT + c];
    __hip_atomic_fetch_add(&agg[(size_t)r * SPLIT + c], v,
                           __ATOMIC_RELAXED, __HIP_MEMORY_SCOPE_AGENT);
}

// ---------------------------------------------------------------------------
// f[c] = mean_n relu( (c<SPLIT ? agg[n,c] : v_s[n,c]) + bias[c] )
// 128 threads/block (one column each), 256 nodes per block, then one
// scaled relaxed float atomic per column into d_out (pre-zeroed).
// ---------------------------------------------------------------------------
__global__ void __launch_bounds__(128) gcn_reduce(
    const float* __restrict__ Vs,
    const float* __restrict__ agg,
    const float* __restrict__ bias,
    float*       __restrict__ out,
    int n_nodes, float inv_n)
{
    const int   c = threadIdx.x;        // 0..127
    const float b = bias[c];
    int start = blockIdx.x * 256;
    int end   = start + 256;
    if (end > n_nodes) end = n_nodes;

    float s = 0.0f;
    if (c < SPLIT) {
        for (int n = start; n < end; ++n) {
            float x = agg[(size_t)n * SPLIT + c] + b;
            s += (x > 0.0f) ? x : 0.0f;
        }
    } else {
        for (int n = start; n < end; ++n) {
            float x = Vs[(size_t)n * FEAT + c] + b;
            s += (x > 0.0f) ? x : 0.0f;
        }
    }
    __hip_atomic_fetch_add(&out[c], s * inv_n,
                           __ATOMIC_RELAXED, __HIP_MEMORY_SCOPE_AGENT);
}

// ---------------------------------------------------------------------------
extern "C" void kernel_launch(void* const* d_in, const int* in_sizes, int n_in,
                              void* d_out, int out_size, void* d_ws, size_t ws_size,
                              hipStream_t stream)
{
    const float* r_s  = (const float*)d_in[0];   // [n_nodes, 128] fp32
    const int*   ei   = (const int*)  d_in[1];   // [2, n_edges]   int
    const float* W    = (const float*)d_in[2];   // [128, 128]     fp32
    const float* bias = (const float*)d_in[3];   // [128]          fp32
    float*       out  = (float*)d_out;           // [128]          fp32

    const int n_nodes = in_sizes[0] / FEAT;      // 50000
    const int n_edges = in_sizes[1] / 2;         // 800000

    float* Vs  = (float*)d_ws;                          // 50000*128 floats
    float* agg = Vs + (size_t)n_nodes * FEAT;           // 50000*12  floats

    // Zero the atomic targets (graph-capturable memset nodes).
    hipMemsetAsync(agg, 0, (size_t)n_nodes * SPLIT * sizeof(float), stream);
    hipMemsetAsync(out, 0, (size_t)out_size * sizeof(float), stream);

    // 1) GEMM: one block per 16-row stripe (50000 = 3125 * 16).
    int gblocks = (n_nodes + TILE_M - 1) / TILE_M;
    gcn_gemm_wmma<<<gblocks, 256, 0, stream>>>(r_s, W, Vs, n_nodes);

    // 2) Edge scatter over the 12 live columns only.
    long long total   = (long long)n_edges * SPLIT;
    int       sblocks = (int)((total + 255) / 256);
    gcn_scatter<<<sblocks, 256, 0, stream>>>(Vs, ei, agg, n_edges);

    // 3) Fused concat + bias + relu + column-mean.
    int rblocks = (n_nodes + 255) / 256;
    gcn_reduce<<<rblocks, 128, 0, stream>>>(Vs, agg, bias, out,
                                            n_nodes, 1.0f / (float)n_nodes);
}